// MiMoV2MoEGate_39462159515918
// MI455X (gfx1250) — compile-verified
//
#include <hip/hip_runtime.h>

typedef __attribute__((ext_vector_type(2))) float v2f;
typedef __attribute__((ext_vector_type(4))) float v4f;
typedef __attribute__((ext_vector_type(8))) float v8f;

#define HIDDEN   4096
#define NEXP     256
#define TOPK     8
#define NGROUP   8
#define GSIZE    32
#define TOPKG    4
#define KCHUNK   64
#define MTOK     32            // tokens per block (two 16-row M-tiles)
#define SCSTRIDE 257           // 256 + pad to dodge bank conflicts

__global__ __launch_bounds__(256) void moe_gate_fused(
    const float* __restrict__ x,       // [n_tok, HIDDEN]
    const float* __restrict__ weight,  // [NEXP, HIDDEN]
    const float* __restrict__ bias,    // [NEXP]
    int*   __restrict__ idx_out,       // [n_tok, TOPK]
    float* __restrict__ w_out,         // [n_tok, TOPK]
    int n_tok)
{
    __shared__ float xs[MTOK * KCHUNK];      // A strip: 32 tokens x KCHUNK (8 KB)
    __shared__ float sc[MTOK * SCSTRIDE];    // sigmoid scores: 32 tokens x 256
    __shared__ float bs[NEXP];               // bias copy

    const int tid  = threadIdx.x;
    const int lane = tid & 31;
    const int wave = tid >> 5;
    const int half = (lane >> 4) & 1;        // 0: lanes 0-15, 1: lanes 16-31
    const int l16  = lane & 15;
    const int m0   = blockIdx.x * MTOK;      // first token of this tile

    bs[tid & (NEXP - 1)] = bias[tid & (NEXP - 1)];

    // Wave w owns expert tiles [32w,32w+16) and [32w+16,32w+32).
    const float* wrow0 = weight + (size_t)(wave * 32 + l16) * HIDDEN + half * 2;
    const float* wrow1 = wrow0 + (size_t)16 * HIDDEN;
    // Two A rows per lane: M-tile 0 (tokens 0..15) and M-tile 1 (tokens 16..31).
    const float* arow0 = xs + l16 * KCHUNK + half * 2;
    const float* arow1 = xs + (16 + l16) * KCHUNK + half * 2;

    v8f acc00 = {};   // M-tile0 x E-tile0
    v8f acc01 = {};   // M-tile0 x E-tile1
    v8f acc10 = {};   // M-tile1 x E-tile0
    v8f acc11 = {};   // M-tile1 x E-tile1

    for (int kb = 0; kb < HIDDEN; kb += KCHUNK) {
        __syncthreads();
        // Cooperative stage of the 32xKCHUNK activation strip (2x float4/thread).
        {
            int f0  = tid * 4;                // first 1024 floats
            int row = f0 >> 6;
            int col = f0 & 63;
            *(v4f*)(xs + f0) = *(const v4f*)(x + (size_t)(m0 + row) * HIDDEN + kb + col);
            int f1 = f0 + 1024;               // second 1024 floats
            *(v4f*)(xs + f1) =
                *(const v4f*)(x + (size_t)(m0 + row + 16) * HIDDEN + kb + col);
        }
        // Prefetch next weight chunk into cache while we compute this one.
        if (kb + KCHUNK < HIDDEN) {
            __builtin_prefetch(wrow0 + kb + KCHUNK, 0, 0);
            __builtin_prefetch(wrow1 + kb + KCHUNK, 0, 0);
        }
        __syncthreads();

        #pragma unroll
        for (int s = 0; s < KCHUNK / 4; ++s) {
            // A fragments: lane (half,l16) holds x[m][k + 2*half + {0,1}]
            v2f a0 = *(const v2f*)(arow0 + 4 * s);
            v2f a1 = *(const v2f*)(arow1 + 4 * s);
            // B fragments: lane holds W[e_base+l16][k + 2*half + {0,1}]
            v2f b0 = *(const v2f*)(wrow0 + kb + 4 * s);
            v2f b1 = *(const v2f*)(wrow1 + kb + 4 * s);
            acc00 = __builtin_amdgcn_wmma_f32_16x16x4_f32(
                false, a0, false, b0, (short)0, acc00, false, false);
            acc01 = __builtin_amdgcn_wmma_f32_16x16x4_f32(
                false, a0, false, b1, (short)0, acc01, false, false);
            acc10 = __builtin_amdgcn_wmma_f32_16x16x4_f32(
                false, a1, false, b0, (short)0, acc10, false, false);
            acc11 = __builtin_amdgcn_wmma_f32_16x16x4_f32(
                false, a1, false, b1, (short)0, acc11, false, false);
        }
    }

    // Sigmoid and scatter scores to LDS.
    // C/D layout: VGPR r, lanes 0-15 -> row r, lanes 16-31 -> row r+8; col = l16.
    #pragma unroll
    for (int r = 0; r < 8; ++r) {
        float v00 = 1.0f / (1.0f + __expf(-acc00[r]));
        float v01 = 1.0f / (1.0f + __expf(-acc01[r]));
        float v10 = 1.0f / (1.0f + __expf(-acc10[r]));
        float v11 = 1.0f / (1.0f + __expf(-acc11[r]));
        int row0 = r + half * 8;          // token row within M-tile 0
        int row1 = row0 + 16;             // token row within M-tile 1
        int c0   = wave * 32 + l16;       // expert column, E-tile 0
        int c1   = c0 + 16;               // expert column, E-tile 1
        sc[row0 * SCSTRIDE + c0] = v00;
        sc[row0 * SCSTRIDE + c1] = v01;
        sc[row1 * SCSTRIDE + c0] = v10;
        sc[row1 * SCSTRIDE + c1] = v11;
    }
    __syncthreads();

    // ---- Fused routing: all 32 lanes of wave 0 each handle one token ----
    if (tid < MTOK) {
        const int t = m0 + tid;
        float* srow = &sc[tid * SCSTRIDE];
        const float NEG = -__builtin_inff();

        // Group scores = sum of top-2 (score + bias) per 32-expert group.
        float gsc[NGROUP];
        #pragma unroll
        for (int g = 0; g < NGROUP; ++g) {
            float m1 = NEG, m2 = NEG;
            for (int j = 0; j < GSIZE; ++j) {
                float v = srow[g * GSIZE + j] + bs[g * GSIZE + j];
                if (v > m1)      { m2 = m1; m1 = v; }
                else if (v > m2) { m2 = v; }
            }
            gsc[g] = m1 + m2;
        }

        // Top-4 groups (ascending scan + strict '>' matches lax.top_k ties).
        unsigned gmask = 0;
        #pragma unroll
        for (int r = 0; r < TOPKG; ++r) {
            float best = NEG; int bg = 0;
            #pragma unroll
            for (int g = 0; g < NGROUP; ++g) {
                if (!((gmask >> g) & 1u) && gsc[g] > best) { best = gsc[g]; bg = g; }
            }
            gmask |= 1u << bg;
        }

        // Top-8 experts among allowed groups; mask chosen by poisoning LDS row.
        int   idxs[TOPK];
        float wts[TOPK];
        float wsum = 0.0f;
        #pragma unroll
        for (int r = 0; r < TOPK; ++r) {
            float best = NEG; int bi = 0;
            for (int g = 0; g < NGROUP; ++g) {
                if (!((gmask >> g) & 1u)) continue;
                for (int j = 0; j < GSIZE; ++j) {
                    int e = g * GSIZE + j;
                    float v = srow[e] + bs[e];
                    if (v > best) { best = v; bi = e; }
                }
            }
            idxs[r] = bi;
            wts[r]  = srow[bi];     // raw sigmoid score (no bias)
            wsum   += srow[bi];
            srow[bi] = NEG;         // exclude from later rounds
        }

        const float scale = 2.5f / (wsum + 1e-20f);
        #pragma unroll
        for (int r = 0; r < TOPK; ++r) {
            idx_out[(size_t)t * TOPK + r] = idxs[r];
            w_out[(size_t)t * TOPK + r]   = wts[r] * scale;
        }
    }
}

extern "C" void kernel_launch(void* const* d_in, const int* in_sizes, int n_in,
                              void* d_out, int out_size, void* d_ws, size_t ws_size,
                              hipStream_t stream) {
    const float* x      = (const float*)d_in[0];   // hidden_states [4,4096,4096] fp32
    const float* weight = (const float*)d_in[1];   // [256,4096] fp32
    const float* bias   = (const float*)d_in[2];   // [256] fp32

    const int n_tok = in_sizes[0] / HIDDEN;        // 16384

    int*   idx_out = (int*)d_out;                            // first n_tok*TOPK ints
    float* w_out   = (float*)d_out + (size_t)n_tok * TOPK;   // then n_tok*TOPK floats

    dim3 grid(n_tok / MTOK), block(256);
    hipLaunchKernelGGL(moe_gate_fused, grid, block, 0, stream,
                       x, weight, bias, idx_out, w_out, n_tok);
}